// NetEncode_61692910240506
// MI455X (gfx1250) — compile-verified
//
#include <hip/hip_runtime.h>

typedef float v2f __attribute__((ext_vector_type(2)));
typedef float v8f __attribute__((ext_vector_type(8)));

#define IN_DIM  256
#define OUT_DIM 64

// ---------------- degree / norm ----------------
__global__ void sg_init_deg(float* __restrict__ deg, int nN) {
    int i = blockIdx.x * blockDim.x + threadIdx.x;
    if (i < nN) deg[i] = 1.0f;                    // self-loop contribution
}

__global__ void sg_count_deg(const int* __restrict__ dst, float* __restrict__ deg, int nE) {
    int e = blockIdx.x * blockDim.x + threadIdx.x;
    if (e < nE) atomicAdd(&deg[dst[e]], 1.0f);
}

__global__ void sg_finish_dinv(float* __restrict__ deg, int nN) {
    int i = blockIdx.x * blockDim.x + threadIdx.x;
    if (i < nN) deg[i] = rsqrtf(deg[i]);          // deg >= 1 always
}

// ---------------- H0 = X @ W^T via V_WMMA_F32_16X16X4_F32 ----------------
// One wave per 16-row M tile, all 4 N-tiles (OUT_DIM=64) per wave.
__global__ __launch_bounds__(256) void sg_gemm_wmma(
    const float* __restrict__ X, const float* __restrict__ W,
    float* __restrict__ H, int nN)
{
    const int lane  = threadIdx.x & 31;
    const int wave  = threadIdx.x >> 5;
    const int tile  = blockIdx.x * 8 + wave;
    const int nTiles = (nN + 15) >> 4;
    if (tile >= nTiles) return;                   // wave-uniform: EXEC stays all-ones
    const int m0   = tile << 4;
    const int hsel = lane >> 4;                   // 0: K pair {0,1}, 1: K pair {2,3}
    const int l15  = lane & 15;

    // A row (clamped so partial tiles never read OOB; stores are guarded below)
    int arow = m0 + l15; if (arow >= nN) arow = nN - 1;
    const float* xrow = X + (size_t)arow * IN_DIM + 2 * hsel;
    // B columns: W[n][k] is exactly B[k][n]; per-lane contiguous float2 in k
    const float* w0 = W + (size_t)(l15 +  0) * IN_DIM + 2 * hsel;
    const float* w1 = W + (size_t)(l15 + 16) * IN_DIM + 2 * hsel;
    const float* w2 = W + (size_t)(l15 + 32) * IN_DIM + 2 * hsel;
    const float* w3 = W + (size_t)(l15 + 48) * IN_DIM + 2 * hsel;

    v8f c0 = {}, c1 = {}, c2 = {}, c3 = {};
#pragma unroll 4
    for (int kk = 0; kk < IN_DIM; kk += 4) {
        v2f a  = *(const v2f*)(xrow + kk);
        v2f b0 = *(const v2f*)(w0 + kk);
        v2f b1 = *(const v2f*)(w1 + kk);
        v2f b2 = *(const v2f*)(w2 + kk);
        v2f b3 = *(const v2f*)(w3 + kk);
        c0 = __builtin_amdgcn_wmma_f32_16x16x4_f32(false, a, false, b0, (short)0, c0, false, false);
        c1 = __builtin_amdgcn_wmma_f32_16x16x4_f32(false, a, false, b1, (short)0, c1, false, false);
        c2 = __builtin_amdgcn_wmma_f32_16x16x4_f32(false, a, false, b2, (short)0, c2, false, false);
        c3 = __builtin_amdgcn_wmma_f32_16x16x4_f32(false, a, false, b3, (short)0, c3, false, false);
    }

    // D layout: VGPR r -> M = r + 8*hsel, N = l15 (+ n0)
#pragma unroll
    for (int r = 0; r < 8; ++r) {
        const int row = m0 + r + (hsel << 3);
        if (row < nN) {
            float* o = H + (size_t)row * OUT_DIM;
            o[ 0 + l15] = c0[r];
            o[16 + l15] = c1[r];
            o[32 + l15] = c2[r];
            o[48 + l15] = c3[r];
        }
    }
}

// ---------------- propagation hop ----------------
// Hout[i,:] = dinv[i]^2 * Hin[i,:]   (self-loop term; also initializes Hout)
__global__ void sg_self_loop(const float* __restrict__ dinv,
                             const float* __restrict__ Hin,
                             float* __restrict__ Hout, int nN)
{
    int tid  = blockIdx.x * blockDim.x + threadIdx.x;
    int node = tid >> 4, q = tid & 15;            // 16 lanes x float4 = 64 dims
    if (node >= nN) return;
    float di = dinv[node];
    float w  = di * di;
    float4 v = ((const float4*)Hin)[(size_t)node * 16 + q];
    float4 o; o.x = w * v.x; o.y = w * v.y; o.z = w * v.z; o.w = w * v.w;
    ((float4*)Hout)[(size_t)node * 16 + q] = o;
}

// Hout[dst,:] += dinv[src]*dinv[dst] * Hin[src,:]
__global__ void sg_edge_scatter(const int* __restrict__ src,
                                const int* __restrict__ dst,
                                const float* __restrict__ dinv,
                                const float* __restrict__ Hin,
                                float* __restrict__ Hout, int nE)
{
    int tid = blockIdx.x * blockDim.x + threadIdx.x;
    int e = tid >> 4, q = tid & 15;
    if (e >= nE) return;
    int s = src[e], d = dst[e];
    float w = dinv[s] * dinv[d];
    float4 v = ((const float4*)(Hin + (size_t)s * OUT_DIM))[q];
    float* o = Hout + (size_t)d * OUT_DIM + (q << 2);
    atomicAdd(o + 0, w * v.x);
    atomicAdd(o + 1, w * v.y);
    atomicAdd(o + 2, w * v.z);
    atomicAdd(o + 3, w * v.w);
}

extern "C" void kernel_launch(void* const* d_in, const int* in_sizes, int n_in,
                              void* d_out, int out_size, void* d_ws, size_t ws_size,
                              hipStream_t stream)
{
    const float* x   = (const float*)d_in[0];
    const int*   ei  = (const int*)  d_in[1];
    const float* W   = (const float*)d_in[2];
    float*       out = (float*)d_out;

    const int nN = in_sizes[0] / IN_DIM;
    const int nE = in_sizes[1] / 2;
    const int* srcp = ei;          // edge_index row 0
    const int* dstp = ei + nE;     // edge_index row 1

    // workspace carve-out (256B aligned): dinv | H0 | H1  (~25.8 MB total)
    char*  ws  = (char*)d_ws;
    size_t off = 0;
    auto carve = [&](size_t bytes) {
        void* p = ws + off;
        off = (off + bytes + 255) & ~(size_t)255;
        return p;
    };
    float* dinv = (float*)carve((size_t)nN * sizeof(float));
    float* H0   = (float*)carve((size_t)nN * OUT_DIM * sizeof(float));
    float* H1   = (float*)carve((size_t)nN * OUT_DIM * sizeof(float));
    (void)ws_size; (void)n_in; (void)out_size;

    const int B = 256;
    // symmetric GCN norm: dinv = rsqrt(1 + indegree)
    sg_init_deg  <<<(nN + B - 1) / B, B, 0, stream>>>(dinv, nN);
    sg_count_deg <<<(nE + B - 1) / B, B, 0, stream>>>(dstp, dinv, nE);
    sg_finish_dinv<<<(nN + B - 1) / B, B, 0, stream>>>(dinv, nN);

    // project first (propagation commutes with the linear layer): H0 = X @ W^T
    const int nTiles = (nN + 15) / 16;
    sg_gemm_wmma<<<(nTiles + 7) / 8, 256, 0, stream>>>(x, W, H0, nN);

    const long thN = (long)nN * 16;
    const long thE = (long)nE * 16;
    // hop 1: H1 = S @ H0
    sg_self_loop   <<<(thN + B - 1) / B, B, 0, stream>>>(dinv, H0, H1, nN);
    sg_edge_scatter<<<(thE + B - 1) / B, B, 0, stream>>>(srcp, dstp, dinv, H0, H1, nE);
    // hop 2: out = S @ H1
    sg_self_loop   <<<(thN + B - 1) / B, B, 0, stream>>>(dinv, H1, out, nN);
    sg_edge_scatter<<<(thE + B - 1) / B, B, 0, stream>>>(srcp, dstp, dinv, H1, out, nE);
}